// DiodeClipper_29798483100262
// MI455X (gfx1250) — compile-verified
//
#include <hip/hip_runtime.h>
#include <math.h>

// Diode clipper: strictly sequential recurrence over T timesteps, B independent
// channels. Latency-bound (48000-step serial Newton chain, only 512-way
// parallel) -> one lane per channel, one wave32 per block so each of the 16
// waves owns its own SIMD. float4 streaming I/O + global_prefetch_b8.

namespace {
// Constants evaluated in double (as Python does), rounded once to f32
// (matching JAX weak-type scalar -> f32 promotion).
constexpr double Kd     = 1.0 / 48000.0;              // timestep
constexpr double OMEGAd = 1.0 / (1000.0 * 3.3e-8);    // 1/(R*C)
constexpr double SINHd  = 2.0 * 2.52e-9 / 3.3e-8;     // 2*I_S/C
constexpr double VTd    = 0.02583;
}

__global__ __launch_bounds__(32)
void diode_clipper_scan(const float* __restrict__ x,
                        const float* __restrict__ h0,
                        float* __restrict__ states,
                        float* __restrict__ h_final,
                        int B, int T)
{
    const int b = blockIdx.x * 32 + threadIdx.x;
    if (b >= B) return;

    const float OMEGA = (float)OMEGAd;
    const float SINHC = (float)SINHd;
    const float VT    = (float)VTd;
    const float invVT = (float)(1.0 / VTd);
    const float halfK = (float)(0.5 * Kd);
    const float Kf    = (float)Kd;
    (void)VT;

    const float* __restrict__ xr = x      + (size_t)b * (size_t)T;
    float*       __restrict__ yr = states + (size_t)b * (size_t)T;

    float h = h0[b];

    const int T4 = T & ~3;
    for (int t0 = 0; t0 < T4; t0 += 4) {
        // Pull the private input stream ahead of the serial Newton chain.
        if ((t0 & 1023) == 0) {
            __builtin_prefetch(xr + t0 + 4096, 0, 1);   // -> global_prefetch_b8
        }
        const float4 xin = *reinterpret_cast<const float4*>(xr + t0);
        float hout[4];
        #pragma unroll
        for (int j = 0; j < 4; ++j) {
            const float xv = (j == 0) ? xin.x : (j == 1) ? xin.y
                           : (j == 2) ? xin.z : xin.w;

            // p = (K/2)*nl(h) - K*(x*OMEGA) - h  at the incoming h (warm start)
            float u  = h * invVT;
            float e  = __expf(u);
            float ei = __expf(-u);
            float sh = 0.5f * (e - ei);
            float nl = OMEGA * h + SINHC * sh;
            float p  = halfK * nl - Kf * (xv * OMEGA) - h;

            // Newton-Raphson; per-lane early exit reproduces the reference's
            // active-mask semantics exactly (step applied, then lane freezes).
            #pragma unroll 1
            for (int it = 0; it < 50; ++it) {
                float uu  = h * invVT;
                float ee  = __expf(uu);
                float eei = __expf(-uu);
                float shh = 0.5f * (ee - eei);
                float chh = 0.5f * (ee + eei);
                float g   = h + halfK * (OMEGA * h + SINHC * shh) + p;
                float dg  = 1.0f + halfK * (OMEGA + SINHC * chh * invVT);
                float step = __fdividef(g, dg);
                h -= step;
                if (fabsf(step) <= 1e-9f) break;
            }
            hout[j] = h;
        }
        float4 ho;
        ho.x = hout[0]; ho.y = hout[1]; ho.z = hout[2]; ho.w = hout[3];
        *reinterpret_cast<float4*>(yr + t0) = ho;
    }

    // Generic tail (T=48000 is divisible by 4; kept for safety).
    for (int t = T4; t < T; ++t) {
        const float xv = xr[t];
        float u  = h * invVT;
        float e  = __expf(u);
        float ei = __expf(-u);
        float sh = 0.5f * (e - ei);
        float nl = OMEGA * h + SINHC * sh;
        float p  = halfK * nl - Kf * (xv * OMEGA) - h;
        #pragma unroll 1
        for (int it = 0; it < 50; ++it) {
            float uu  = h * invVT;
            float ee  = __expf(uu);
            float eei = __expf(-uu);
            float shh = 0.5f * (ee - eei);
            float chh = 0.5f * (ee + eei);
            float g   = h + halfK * (OMEGA * h + SINHC * shh) + p;
            float dg  = 1.0f + halfK * (OMEGA + SINHC * chh * invVT);
            float step = __fdividef(g, dg);
            h -= step;
            if (fabsf(step) <= 1e-9f) break;
        }
        yr[t] = h;
    }

    h_final[b] = h;
}

extern "C" void kernel_launch(void* const* d_in, const int* in_sizes, int n_in,
                              void* d_out, int out_size, void* d_ws, size_t ws_size,
                              hipStream_t stream) {
    (void)n_in; (void)out_size; (void)d_ws; (void)ws_size;

    const float* x  = (const float*)d_in[0];   // [B, T, 1]
    const float* h0 = (const float*)d_in[1];   // [B, 1]
    float* out = (float*)d_out;                // states [B*T] then h_final [B]

    const int B = in_sizes[1];
    const int T = in_sizes[0] / B;

    float* states  = out;
    float* h_final = out + (size_t)B * (size_t)T;

    dim3 block(32);                 // one wave32 per block -> private SIMD
    dim3 grid((B + 31) / 32);       // 16 blocks for B=512
    hipLaunchKernelGGL(diode_clipper_scan, grid, block, 0, stream,
                       x, h0, states, h_final, B, T);
}